// CompGCNLayer_74431783240016
// MI455X (gfx1250) — compile-verified
//
#include <hip/hip_runtime.h>
#include <hip/hip_bf16.h>

typedef __attribute__((ext_vector_type(2))) float v2f;
typedef __attribute__((ext_vector_type(8))) float v8f;

#define DIM 128

// ---------------------------------------------------------------------------
// Zero a float buffer (graph-capture-safe alternative to hipMemsetAsync).
// ---------------------------------------------------------------------------
__global__ void zero_f32(float* __restrict__ p, long long n4) {
    long long i = (long long)blockIdx.x * blockDim.x + threadIdx.x;
    if (i < n4) {
        float4 z = {0.f, 0.f, 0.f, 0.f};
        ((float4*)p)[i] = z;
    }
}

// ---------------------------------------------------------------------------
// C[M,128] = A[M,128] @ W[128,128] using V_WMMA_F32_16X16X4_F32.
// Block = 256 threads = 8 wave32; wave w computes the 16x16 tile at
// (m0 = 16*blockIdx.x, n0 = 16*w).  K loop: 32 steps of K=4.
//
// Lane layout per CDNA5 ISA 7.12.2 (32-bit A 16x4):
//   lanes 0-15 : row m = lane,     vgpr0 = K0, vgpr1 = K1
//   lanes 16-31: row m = lane-16,  vgpr0 = K2, vgpr1 = K3
// B (4x16) striped across lanes within a VGPR, same half-wave K split.
// C/D: 8 VGPRs, lanes 0-15 -> M=0..7, lanes 16-31 -> M=8..15, N = lane%16.
//
// Ragged M edge handled by CLAMPING the load row (no EXEC divergence in the
// hot loop — keeps the load stream branch-free); stores remain guarded.
// ---------------------------------------------------------------------------
__global__ void gemm128_wmma_f32(const float* __restrict__ A,
                                 const float* __restrict__ W,
                                 float* __restrict__ C, int M) {
    const int wave = threadIdx.x >> 5;
    const int lane = threadIdx.x & 31;
    const int half = lane & 15;
    const bool hi  = lane >= 16;
    const int koff = hi ? 2 : 0;

    const int m0 = blockIdx.x * 16;
    const int n0 = wave * 16;

    const int mrow  = m0 + half;                    // A row fed by this lane
    const int mload = (mrow < M) ? mrow : (M - 1);  // clamp: always in-bounds
    const int ncol  = n0 + half;                    // B/C column for this lane

    v8f acc = {};
    const float* __restrict__ arow = A + (long long)mload * DIM;
    const float* __restrict__ wcol = W + (long long)koff * DIM + ncol;

#pragma unroll
    for (int k0 = 0; k0 < DIM; k0 += 4) {
        v2f a = *(const v2f*)(arow + k0 + koff);    // 8B aligned (koff even)
        v2f b;
        b.x = wcol[(long long)k0 * DIM];
        b.y = wcol[(long long)k0 * DIM + DIM];
        acc = __builtin_amdgcn_wmma_f32_16x16x4_f32(
                  /*neg_a=*/false, a, /*neg_b=*/false, b,
                  /*c_mod=*/(short)0, acc,
                  /*reuse_a=*/false, /*reuse_b=*/false);
    }

    const int rbase = m0 + (hi ? 8 : 0);
#pragma unroll
    for (int i = 0; i < 8; ++i) {
        const int row = rbase + i;
        if (row < M) C[(long long)row * DIM + ncol] = acc[i];
    }
}

// ---------------------------------------------------------------------------
// Edge scatter: one wave per edge.  Lane l handles elements [4l, 4l+4).
// msg = hw[src] - rw[etype]; atomically accumulated into agg[dst].
// hw / rw / agg are all L2-resident (51.2MB + 256KB + 51.2MB << 192MB L2),
// so the gathers and f32 atomics run at L2 rates, not HBM.
// readfirstlane makes the (wave-uniform) row bases scalar so the gathers
// and atomics use the SGPR-base + lane-offset addressing form.
// ---------------------------------------------------------------------------
__global__ void edge_scatter(const float* __restrict__ hw,
                             const float* __restrict__ rw,
                             const int* __restrict__ src,
                             const int* __restrict__ dst,
                             const int* __restrict__ etype,
                             float* __restrict__ agg, int E) {
    const int edge = blockIdx.x * (blockDim.x >> 5) + (threadIdx.x >> 5);
    const int lane = threadIdx.x & 31;
    if (edge >= E) return;

    const int s = __builtin_amdgcn_readfirstlane(src[edge]);
    const int d = __builtin_amdgcn_readfirstlane(dst[edge]);
    const int r = __builtin_amdgcn_readfirstlane(etype[edge]);

    const float4 hv = *(const float4*)(hw + (long long)s * DIM + lane * 4);
    const float4 rv = *(const float4*)(rw + (long long)r * DIM + lane * 4);

    float* base = agg + (long long)d * DIM + lane * 4;
    __hip_atomic_fetch_add(base + 0, hv.x - rv.x, __ATOMIC_RELAXED, __HIP_MEMORY_SCOPE_AGENT);
    __hip_atomic_fetch_add(base + 1, hv.y - rv.y, __ATOMIC_RELAXED, __HIP_MEMORY_SCOPE_AGENT);
    __hip_atomic_fetch_add(base + 2, hv.z - rv.z, __ATOMIC_RELAXED, __HIP_MEMORY_SCOPE_AGENT);
    __hip_atomic_fetch_add(base + 3, hv.w - rv.w, __ATOMIC_RELAXED, __HIP_MEMORY_SCOPE_AGENT);
}

// ---------------------------------------------------------------------------
// out = relu(agg * norm + out)   (out already holds the self-loop GEMM)
// One thread per float4 (N*32 groups).
// ---------------------------------------------------------------------------
__global__ void finalize(const float* __restrict__ agg,
                         const float* __restrict__ norm,
                         float* __restrict__ out, long long n4) {
    long long i = (long long)blockIdx.x * blockDim.x + threadIdx.x;
    if (i >= n4) return;
    const int node = (int)(i >> 5);          // 32 float4 per node (DIM=128)
    const float nm = norm[node];
    const float4 a = ((const float4*)agg)[i];
    float4 o = ((float4*)out)[i];
    o.x = fmaxf(a.x * nm + o.x, 0.f);
    o.y = fmaxf(a.y * nm + o.y, 0.f);
    o.z = fmaxf(a.z * nm + o.z, 0.f);
    o.w = fmaxf(a.w * nm + o.w, 0.f);
    ((float4*)out)[i] = o;
}

// ---------------------------------------------------------------------------
// Launch
// ---------------------------------------------------------------------------
extern "C" void kernel_launch(void* const* d_in, const int* in_sizes, int n_in,
                              void* d_out, int out_size, void* d_ws, size_t ws_size,
                              hipStream_t stream) {
    const float* h      = (const float*)d_in[0];   // [N,128]
    const float* norm   = (const float*)d_in[1];   // [N,1]
    const float* rel    = (const float*)d_in[2];   // [R,128]
    const float* Wn     = (const float*)d_in[3];   // [128,128]
    const float* Wl     = (const float*)d_in[4];   // [128,128]
    const int*   src    = (const int*)d_in[5];     // [E]
    const int*   dst    = (const int*)d_in[6];     // [E]
    const int*   etype  = (const int*)d_in[7];     // [E]
    float*       out    = (float*)d_out;           // [N,128]

    const int N = in_sizes[0] / DIM;
    const int R = in_sizes[2] / DIM;
    const int E = in_sizes[5];

    // Workspace layout (floats): hw[N*128] | rw[Rpad*128] | agg[N*128]
    const int Rpad = (R + 15) & ~15;
    float* hw  = (float*)d_ws;
    float* rw  = hw + (long long)N * DIM;
    float* agg = rw + (long long)Rpad * DIM;

    // 1) agg = 0
    {
        long long n4 = (long long)N * DIM / 4;
        int blocks = (int)((n4 + 255) / 256);
        zero_f32<<<blocks, 256, 0, stream>>>(agg, n4);
    }

    // 2) GEMMs (WMMA f32 16x16x4): hw = h@Wn, rw = rel@Wn, out = h@Wl
    {
        int mtN = (N + 15) / 16;
        int mtR = (R + 15) / 16;
        gemm128_wmma_f32<<<mtN, 256, 0, stream>>>(h,   Wn, hw,  N);
        gemm128_wmma_f32<<<mtR, 256, 0, stream>>>(rel, Wn, rw,  R);
        gemm128_wmma_f32<<<mtN, 256, 0, stream>>>(h,   Wl, out, N);
    }

    // 3) Edge gather / scatter-add (wave per edge, 8 edges per block)
    {
        int blocks = (E + 7) / 8;
        edge_scatter<<<blocks, 256, 0, stream>>>(hw, rw, src, dst, etype, agg, E);
    }

    // 4) out = relu(agg*norm + out)
    {
        long long n4 = (long long)N * DIM / 4;
        int blocks = (int)((n4 + 255) / 256);
        finalize<<<blocks, 256, 0, stream>>>(agg, norm, out, n4);
    }
}